// Encoder_5136780886772
// MI455X (gfx1250) — compile-verified
//
#include <hip/hip_runtime.h>

// Fused transformer encoder layer for MI455X (gfx1250), wave32, fp32 WMMA.
// Persistent kernel: 1024 WGs (128 thr = 4 waves) loop over 16-batch tiles.
// Weights staged in LDS once per WG. Per tile: QKV proj (WMMA, dual-acc) ->
// attention (VALU) -> Wo (WMMA) -> +residual -> LN1 -> fused FFN (WMMA,
// 2 hidden tiles in flight, hidden never leaves LDS) -> +residual -> LN2.
// Pad-column stores go to an LDS dump region (address select, no exec-mask
// churn); hot activation buffers are declared first so DS immediate offsets
// fold. Only x/out touch HBM; next tile's x is prefetched during compute.

namespace {
constexpr int kB    = 65536;
constexpr int kS    = 9;
constexpr int kD    = 20;
constexpr int kH    = 4;
constexpr int kDH   = 5;
constexpr int kDFF  = 512;
constexpr int kNB   = 16;               // batches per tile
constexpr int kRows = kNB * kS;         // 144 rows
constexpr int kNT   = kRows / 16;       // 9 M-tiles of 16 rows (exact)
constexpr int kTiles = kB / kNB;        // 4096 tiles
constexpr int kGrid  = 1024;            // persistent workgroups (~4 tiles each)
constexpr int kWaves = 4;
constexpr int kBlock = kWaves * 32;     // 128 threads (wave32)
constexpr float kEps = 1e-5f;
}

typedef float v2f __attribute__((ext_vector_type(2)));
typedef float v8f __attribute__((ext_vector_type(8)));

static __device__ __forceinline__ v8f wmma4(v2f a, v2f b, v8f c) {
  // V_WMMA_F32_16X16X4_F32: D = A(16x4) x B(4x16) + C(16x16), all fp32
  return __builtin_amdgcn_wmma_f32_16x16x4_f32(false, a, false, b, (short)0, c,
                                               false, false);
}

// A operand (16x4): lanes 0-15 rows M with K = k0+{0,1}; lanes 16-31 K = k0+{2,3}
static __device__ __forceinline__ v2f ldA(const float* rowbase, int stride,
                                          int k0, int lane) {
  const int half = lane >> 4;
  const int m = lane & 15;
  const float* p = rowbase + m * stride + k0 + 2 * half;
  v2f r; r.x = p[0]; r.y = p[1];
  return r;
}

// B operand (4x16) from row-major W[K][N], columns n0..n0+15 (clamped for N<16 pad)
static __device__ __forceinline__ v2f ldB(const float* W, int N, int k0, int n0,
                                          int lane) {
  const int half = lane >> 4;
  int n = n0 + (lane & 15);
  if (n > N - 1) n = N - 1;            // junk columns, never kept
  const float* p = W + (k0 + 2 * half) * N + n;
  v2f r; r.x = p[0]; r.y = p[N];
  return r;
}

__global__ __launch_bounds__(kBlock)
void encoder_fused(const float* __restrict__ X,
                   const float* __restrict__ Wq, const float* __restrict__ bq,
                   const float* __restrict__ Wk, const float* __restrict__ bk,
                   const float* __restrict__ Wv, const float* __restrict__ bv,
                   const float* __restrict__ Wo, const float* __restrict__ bo,
                   const float* __restrict__ g1, const float* __restrict__ b1,
                   const float* __restrict__ W1, const float* __restrict__ W2,
                   const float* __restrict__ g2, const float* __restrict__ b2,
                   float* __restrict__ out) {
  // Hot activation buffers first: keep their DS addresses < 64KB so the
  // 16-bit DS immediate-offset field can absorb the base + j*stride deltas.
  __shared__ float qs[kRows * kD];            // q, later y (post-LN1)
  __shared__ float hstage[kWaves * 16 * 32];  // per-wave relu(h) C->A relayout
  __shared__ float ms[kRows * kD];            // merged ctx, later pre-LN2 out
  __shared__ float xs[kRows * kD];            // input tile (kept for residual)
  __shared__ float ks[kRows * kD];
  __shared__ float vs[kRows * kD];
  __shared__ float sVec[8 * kD];              // bq,bk,bv,bo,g1,b1,g2,b2
  __shared__ float sDump[16 * kD + 16];       // junk sink for pad-column stores
  __shared__ float sWq[kD * kD], sWk[kD * kD], sWv[kD * kD], sWo[kD * kD];
  __shared__ float sW1[kD * kDFF];            // 40 KB
  __shared__ float sW2[kDFF * kD];            // 40 KB

  const int tid  = threadIdx.x;
  const int lane = tid & 31;
  const int wave = tid >> 5;
  const int half = lane >> 4;
  const int cl   = lane & 15;
  const bool live1 = (cl < 4);                // pad-tile column valid?
  const int col1r = 16 + (cl & 3);            // clamped in-bounds pad column
  float* const dumpBase = sDump + (8 * half) * kD + cl;

  // ---- stage weights once per (persistent) workgroup ----
  for (int i = tid; i < kD * kDFF; i += kBlock) sW1[i] = W1[i];
  for (int i = tid; i < kDFF * kD; i += kBlock) sW2[i] = W2[i];
  for (int i = tid; i < kD * kD; i += kBlock) {
    sWq[i] = Wq[i]; sWk[i] = Wk[i]; sWv[i] = Wv[i]; sWo[i] = Wo[i];
  }
  if (tid < kD) {
    sVec[0 * kD + tid] = bq[tid]; sVec[1 * kD + tid] = bk[tid];
    sVec[2 * kD + tid] = bv[tid]; sVec[3 * kD + tid] = bo[tid];
    sVec[4 * kD + tid] = g1[tid]; sVec[5 * kD + tid] = b1[tid];
    sVec[6 * kD + tid] = g2[tid]; sVec[7 * kD + tid] = b2[tid];
  }

  for (int tile = blockIdx.x; tile < kTiles; tile += kGrid) {
    // ---- stage activations; prefetch next tile while this one computes ----
    const float* xg = X + (size_t)tile * kRows * kD;
    for (int i = tid; i < kRows * kD; i += kBlock) xs[i] = xg[i];
    if (tile + kGrid < kTiles && tid < 90) {
      const float* nxt = X + (size_t)(tile + kGrid) * kRows * kD;
      __builtin_prefetch(nxt + tid * 32, 0, 3);  // global_prefetch_b8, 128B/line
    }
    __syncthreads();   // also covers weight staging on first iteration

    // ---- Q/K/V projections: [16,20] @ [20,20] + bias, dual N-tile WMMA ----
    for (int t = wave; t < kNT; t += kWaves) {
      const float* arow = xs + t * 16 * kD;
      const int rbase = (t * 16 + 8 * half) * kD;
      for (int p = 0; p < 3; ++p) {
        const float* W = (p == 0) ? sWq : (p == 1) ? sWk : sWv;
        float* dst = (p == 0) ? qs : (p == 1) ? ks : vs;
        const float* bias = sVec + p * kD;
        const float bb0 = bias[cl];
        const float bb1 = bias[col1r];          // junk for dead lanes
        v8f c0, c1;
        #pragma unroll
        for (int j = 0; j < 8; ++j) { c0[j] = bb0; c1[j] = bb1; }
        #pragma unroll
        for (int k0 = 0; k0 < kD; k0 += 4) {
          const v2f a = ldA(arow, kD, k0, lane);
          c0 = wmma4(a, ldB(W, kD, k0, 0, lane), c0);
          c1 = wmma4(a, ldB(W, kD, k0, 16, lane), c1);
        }
        float* d0 = dst + rbase + cl;
        float* d1 = live1 ? (dst + rbase + 16 + cl) : dumpBase;
        #pragma unroll
        for (int j = 0; j < 8; ++j) {
          d0[j * kD] = c0[j];
          d1[j * kD] = c1[j];
        }
      }
    }
    __syncthreads();

    // ---- attention: per (batch, head, query-row) task; S=9, dh=5 ----
    for (int task = tid; task < kNB * kH * kS; task += kBlock) {
      const int b = task / (kH * kS);
      const int r = task % (kH * kS);
      const int h = r / kS;
      const int q = r % kS;
      const float* qv = qs + (b * kS + q) * kD + h * kDH;
      float a[kS];
      float mx = -1e30f;
      #pragma unroll
      for (int j = 0; j < kS; ++j) {
        const float* kv = ks + (b * kS + j) * kD + h * kDH;
        float d = 0.f;
        #pragma unroll
        for (int e = 0; e < kDH; ++e) d += qv[e] * kv[e];
        d *= (1.f / 3.f);               // faithful hardcoded /3
        a[j] = d;
        mx = fmaxf(mx, d);
      }
      float sum = 0.f;
      #pragma unroll
      for (int j = 0; j < kS; ++j) { a[j] = __expf(a[j] - mx); sum += a[j]; }
      const float inv = 1.f / sum;
      float ctx[kDH] = {0.f, 0.f, 0.f, 0.f, 0.f};
      #pragma unroll
      for (int j = 0; j < kS; ++j) {
        const float w = a[j] * inv;
        const float* vv = vs + (b * kS + j) * kD + h * kDH;
        #pragma unroll
        for (int e = 0; e < kDH; ++e) ctx[e] += w * vv[e];
      }
      // faithful quirk: ctx.transpose(0,1,3,2).reshape(B,9,20)
      #pragma unroll
      for (int e = 0; e < kDH; ++e) {
        const int f = (h * kDH + e) * kS + q;
        ms[(b * kS + f / kD) * kD + (f % kD)] = ctx[e];
      }
    }
    __syncthreads();

    // ---- output projection + residual: qs <- merged @ Wo + bo + x ----
    for (int t = wave; t < kNT; t += kWaves) {
      const float* arow = ms + t * 16 * kD;
      const int rbase = (t * 16 + 8 * half) * kD;
      const float bb0 = sVec[3 * kD + cl];
      const float bb1 = sVec[3 * kD + col1r];
      v8f c0, c1;
      #pragma unroll
      for (int j = 0; j < 8; ++j) { c0[j] = bb0; c1[j] = bb1; }
      #pragma unroll
      for (int k0 = 0; k0 < kD; k0 += 4) {
        const v2f a = ldA(arow, kD, k0, lane);
        c0 = wmma4(a, ldB(sWo, kD, k0, 0, lane), c0);
        c1 = wmma4(a, ldB(sWo, kD, k0, 16, lane), c1);
      }
      const float* r0 = xs + rbase + cl;
      const float* r1 = xs + rbase + col1r;     // clamped: junk for dead lanes
      float* d0 = qs + rbase + cl;
      float* d1 = live1 ? (qs + rbase + 16 + cl) : dumpBase;
      #pragma unroll
      for (int j = 0; j < 8; ++j) {
        d0[j * kD] = c0[j] + r0[j * kD];
        d1[j * kD] = c1[j] + r1[j * kD];
      }
    }
    __syncthreads();

    // ---- LayerNorm 1 (in place on qs) ----
    for (int row = tid; row < kRows; row += kBlock) {
      float* y = qs + row * kD;
      float m = 0.f;
      #pragma unroll
      for (int d = 0; d < kD; ++d) m += y[d];
      m *= (1.f / kD);
      float v = 0.f;
      #pragma unroll
      for (int d = 0; d < kD; ++d) { const float u = y[d] - m; v += u * u; }
      const float inv = rsqrtf(v * (1.f / kD) + kEps);
      #pragma unroll
      for (int d = 0; d < kD; ++d)
        y[d] = (y[d] - m) * inv * sVec[4 * kD + d] + sVec[5 * kD + d];
    }
    __syncthreads();

    // ---- fused FFN: ms = qs + relu(qs @ W1) @ W2 ----
    // Two hidden 16-tiles in flight for WMMA ILP; hidden stays in LDS.
    for (int t = wave; t < kNT; t += kWaves) {
      const float* arow = qs + t * 16 * kD;
      float* hs = hstage + wave * (16 * 32);   // wave-private [16][32]
      v8f acc0, acc1;
      #pragma unroll
      for (int j = 0; j < 8; ++j) { acc0[j] = 0.f; acc1[j] = 0.f; }
      for (int nh = 0; nh < kDFF / 16; nh += 2) {
        const int hn0 = nh * 16;
        v8f hc0, hc1;
        #pragma unroll
        for (int j = 0; j < 8; ++j) { hc0[j] = 0.f; hc1[j] = 0.f; }
        #pragma unroll
        for (int k0 = 0; k0 < kD; k0 += 4) {
          const v2f a = ldA(arow, kD, k0, lane);
          hc0 = wmma4(a, ldB(sW1, kDFF, k0, hn0, lane), hc0);
          hc1 = wmma4(a, ldB(sW1, kDFF, k0, hn0 + 16, lane), hc1);
        }
        // ReLU + C-layout -> row-major staging (LDS in-order within a wave)
        float* hw = hs + (8 * half) * 32 + cl;
        #pragma unroll
        for (int j = 0; j < 8; ++j) {
          hw[j * 32]      = fmaxf(hc0[j], 0.f);
          hw[j * 32 + 16] = fmaxf(hc1[j], 0.f);
        }
        #pragma unroll
        for (int kk = 0; kk < 32; kk += 4) {
          const v2f a = ldA(hs, 32, kk, lane);
          acc0 = wmma4(a, ldB(sW2, kD, hn0 + kk, 0, lane), acc0);
          acc1 = wmma4(a, ldB(sW2, kD, hn0 + kk, 16, lane), acc1);
        }
      }
      const int rbase = (t * 16 + 8 * half) * kD;
      const float* r0 = qs + rbase + cl;
      const float* r1 = qs + rbase + col1r;     // clamped: junk for dead lanes
      float* d0 = ms + rbase + cl;
      float* d1 = live1 ? (ms + rbase + 16 + cl) : dumpBase;
      #pragma unroll
      for (int j = 0; j < 8; ++j) {
        d0[j * kD] = acc0[j] + r0[j * kD];
        d1[j * kD] = acc1[j] + r1[j * kD];
      }
    }
    __syncthreads();

    // ---- LayerNorm 2 (in place on ms) ----
    for (int row = tid; row < kRows; row += kBlock) {
      float* y = ms + row * kD;
      float m = 0.f;
      #pragma unroll
      for (int d = 0; d < kD; ++d) m += y[d];
      m *= (1.f / kD);
      float v = 0.f;
      #pragma unroll
      for (int d = 0; d < kD; ++d) { const float u = y[d] - m; v += u * u; }
      const float inv = rsqrtf(v * (1.f / kD) + kEps);
      #pragma unroll
      for (int d = 0; d < kD; ++d)
        y[d] = (y[d] - m) * inv * sVec[6 * kD + d] + sVec[7 * kD + d];
    }
    __syncthreads();

    // ---- contiguous store (next iteration's first conflicting LDS write
    //      is fenced by the in-loop __syncthreads after the xs staging) ----
    float* og = out + (size_t)tile * kRows * kD;
    for (int i = tid; i < kRows * kD; i += kBlock) og[i] = ms[i];
  }
}

extern "C" void kernel_launch(void* const* d_in, const int* in_sizes, int n_in,
                              void* d_out, int out_size, void* d_ws, size_t ws_size,
                              hipStream_t stream) {
  (void)in_sizes; (void)n_in; (void)out_size; (void)d_ws; (void)ws_size;
  const float* X  = (const float*)d_in[0];
  const float* Wq = (const float*)d_in[1];
  const float* bq = (const float*)d_in[2];
  const float* Wk = (const float*)d_in[3];
  const float* bk = (const float*)d_in[4];
  const float* Wv = (const float*)d_in[5];
  const float* bv = (const float*)d_in[6];
  const float* Wo = (const float*)d_in[7];
  const float* bo = (const float*)d_in[8];
  const float* g1 = (const float*)d_in[9];
  const float* b1 = (const float*)d_in[10];
  const float* W1 = (const float*)d_in[11];
  const float* W2 = (const float*)d_in[12];
  const float* g2 = (const float*)d_in[13];
  const float* b2 = (const float*)d_in[14];

  dim3 grid(kGrid);      // persistent: 1024 WGs x ~4 tiles each
  dim3 block(kBlock);    // 128 threads = 4 waves (wave32)
  encoder_fused<<<grid, block, 0, stream>>>(X, Wq, bq, Wk, bk, Wv, bv, Wo, bo,
                                            g1, b1, W1, W2, g2, b2,
                                            (float*)d_out);
}